// CYBORGCrossEntropyLossXReactionTime_77189152244381
// MI455X (gfx1250) — compile-verified
//
#include <hip/hip_runtime.h>
#include <math.h>

// ---------------------------------------------------------------------------
// CDNA5 / gfx1250: wave32. Wave-level float sum via V_WMMA_F32_16X16X4_F32:
// A = lane values (16x4, K=1,3 zero-padded), B = ones(4x16)  =>
// D[m][n] = v_m + v_{m+16}. Each lane holds 8 of the 16 row results; summing
// its 8 accumulator regs + one xor-16 shuffle yields the full 32-lane sum.
// Exact in f32 (multiplies by 1.0 are exact).
// ---------------------------------------------------------------------------
typedef __attribute__((ext_vector_type(2))) float v2f;
typedef __attribute__((ext_vector_type(8))) float v8f;

__device__ __forceinline__ float wave_sum(float v) {
#if defined(__gfx1250__) && __has_builtin(__builtin_amdgcn_wmma_f32_16x16x4_f32)
  v2f a; a[0] = v;    a[1] = 0.0f;
  v2f b; b[0] = 1.0f; b[1] = 1.0f;
  v8f c = {};
  v8f d = __builtin_amdgcn_wmma_f32_16x16x4_f32(
      /*neg_a=*/false, a, /*neg_b=*/false, b,
      /*c_mod=*/(short)0, c, /*reuse_a=*/false, /*reuse_b=*/false);
  float s = ((d[0] + d[1]) + (d[2] + d[3])) + ((d[4] + d[5]) + (d[6] + d[7]));
  s += __shfl_xor(s, 16, 32);
  return s;
#else
  for (int o = 16; o > 0; o >>= 1) v += __shfl_xor(v, o, 32);
  return v;
#endif
}

__device__ __forceinline__ float wave_min(float v) {
  for (int o = 16; o > 0; o >>= 1) v = fminf(v, __shfl_xor(v, o, 32));
  return v;
}
__device__ __forceinline__ float wave_max(float v) {
  for (int o = 16; o > 0; o >>= 1) v = fmaxf(v, __shfl_xor(v, o, 32));
  return v;
}

// ---------------------------------------------------------------------------
// Pass 1: global min/max of cams and of channel-mean(annotations).
// float4 streaming (global_load_b128). HW % 4 == 0 so a float4 never crosses
// a sample boundary. ann layout [B,3,H,W]: 3 strided coalesced loads / pixel4.
// No per-element divisions: mean uses *(1/3).
// ---------------------------------------------------------------------------
__global__ void k_pass1(const float4* __restrict__ cams4,
                        const float4* __restrict__ ann4,
                        float* __restrict__ cminP, float* __restrict__ cmaxP,
                        float* __restrict__ aminP, float* __restrict__ amaxP,
                        int n4, int hw4) {
  const float third = 1.0f / 3.0f;
  float cmn = INFINITY, cmx = -INFINITY, amn = INFINITY, amx = -INFINITY;
  int stride = gridDim.x * blockDim.x;
  for (int j = blockIdx.x * blockDim.x + threadIdx.x; j < n4; j += stride) {
    float4 cv = cams4[j];
    cmn = fminf(cmn, fminf(fminf(cv.x, cv.y), fminf(cv.z, cv.w)));
    cmx = fmaxf(cmx, fmaxf(fmaxf(cv.x, cv.y), fmaxf(cv.z, cv.w)));
    int b  = j / hw4;
    int p4 = j - b * hw4;
    const float4* base = ann4 + (size_t)b * 3 * hw4 + p4;
    float4 a0 = base[0];
    float4 a1 = base[hw4];
    float4 a2 = base[2 * hw4];
    float m0 = (a0.x + a1.x + a2.x) * third;
    float m1 = (a0.y + a1.y + a2.y) * third;
    float m2 = (a0.z + a1.z + a2.z) * third;
    float m3 = (a0.w + a1.w + a2.w) * third;
    amn = fminf(amn, fminf(fminf(m0, m1), fminf(m2, m3)));
    amx = fmaxf(amx, fmaxf(fmaxf(m0, m1), fmaxf(m2, m3)));
  }
  cmn = wave_min(cmn); cmx = wave_max(cmx);
  amn = wave_min(amn); amx = wave_max(amx);
  __shared__ float sred[4][8];
  int lane = threadIdx.x & 31, wid = threadIdx.x >> 5, nw = blockDim.x >> 5;
  if (lane == 0) { sred[0][wid] = cmn; sred[1][wid] = cmx;
                   sred[2][wid] = amn; sred[3][wid] = amx; }
  __syncthreads();
  if (threadIdx.x == 0) {
    float a = sred[0][0], bq = sred[1][0], c = sred[2][0], d = sred[3][0];
    for (int w = 1; w < nw; ++w) {
      a = fminf(a, sred[0][w]); bq = fmaxf(bq, sred[1][w]);
      c = fminf(c, sred[2][w]); d  = fmaxf(d,  sred[3][w]);
    }
    cminP[blockIdx.x] = a; cmaxP[blockIdx.x] = bq;
    aminP[blockIdx.x] = c; amaxP[blockIdx.x] = d;
  }
}

// Reduce block partials -> 4 scalars: [cmin, cmax, amin, amax]
__global__ void k_reduce1(const float* __restrict__ cminP, const float* __restrict__ cmaxP,
                          const float* __restrict__ aminP, const float* __restrict__ amaxP,
                          float* __restrict__ scal, int n) {
  float cmn = INFINITY, cmx = -INFINITY, amn = INFINITY, amx = -INFINITY;
  for (int i = threadIdx.x; i < n; i += blockDim.x) {
    cmn = fminf(cmn, cminP[i]); cmx = fmaxf(cmx, cmaxP[i]);
    amn = fminf(amn, aminP[i]); amx = fmaxf(amx, amaxP[i]);
  }
  cmn = wave_min(cmn); cmx = wave_max(cmx);
  amn = wave_min(amn); amx = wave_max(amx);
  __shared__ float sred[4][8];
  int lane = threadIdx.x & 31, wid = threadIdx.x >> 5, nw = blockDim.x >> 5;
  if (lane == 0) { sred[0][wid] = cmn; sred[1][wid] = cmx;
                   sred[2][wid] = amn; sred[3][wid] = amx; }
  __syncthreads();
  if (threadIdx.x == 0) {
    float a = sred[0][0], bq = sred[1][0], c = sred[2][0], d = sred[3][0];
    for (int w = 1; w < nw; ++w) {
      a = fminf(a, sred[0][w]); bq = fmaxf(bq, sred[1][w]);
      c = fminf(c, sred[2][w]); d  = fmaxf(d,  sred[3][w]);
    }
    scal[0] = a; scal[1] = bq; scal[2] = c; scal[3] = d;
  }
}

// ---------------------------------------------------------------------------
// Pass 2: sum of squared diffs of the two normalized maps.
// Re-reads cams+annotations (103 MB) — fits the 192 MB L2, so this pass is
// expected to be served largely from L2. Normalization via hoisted
// reciprocals (2 real divisions per thread total), so the stream loop is
// sub/mul/fma only and stays memory-bound. Block sum via WMMA wave reduce.
// ---------------------------------------------------------------------------
__global__ void k_pass2(const float4* __restrict__ cams4,
                        const float4* __restrict__ ann4,
                        const float* __restrict__ scal,
                        float* __restrict__ sseP, int n4, int hw4) {
  const float third = 1.0f / 3.0f;
  float cmin = scal[0], amin = scal[2];
  float sc = 1.0f / (scal[1] - scal[0]);   // 1 / (cmax - cmin)
  float sa = 1.0f / (scal[3] - scal[2]);   // 1 / (amax - amin)
  float acc = 0.0f;
  int stride = gridDim.x * blockDim.x;
  for (int j = blockIdx.x * blockDim.x + threadIdx.x; j < n4; j += stride) {
    float4 cv = cams4[j];
    int b  = j / hw4;
    int p4 = j - b * hw4;
    const float4* base = ann4 + (size_t)b * 3 * hw4 + p4;
    float4 a0 = base[0];
    float4 a1 = base[hw4];
    float4 a2 = base[2 * hw4];
    float m0 = (a0.x + a1.x + a2.x) * third;
    float m1 = (a0.y + a1.y + a2.y) * third;
    float m2 = (a0.z + a1.z + a2.z) * third;
    float m3 = (a0.w + a1.w + a2.w) * third;
    float d0 = (cv.x - cmin) * sc - (m0 - amin) * sa;
    float d1 = (cv.y - cmin) * sc - (m1 - amin) * sa;
    float d2 = (cv.z - cmin) * sc - (m2 - amin) * sa;
    float d3 = (cv.w - cmin) * sc - (m3 - amin) * sa;
    acc += (d0 * d0 + d1 * d1) + (d2 * d2 + d3 * d3);
  }
  float s = wave_sum(acc);   // v_wmma_f32_16x16x4_f32 path
  __shared__ float ssum[8];
  int lane = threadIdx.x & 31, wid = threadIdx.x >> 5, nw = blockDim.x >> 5;
  if (lane == 0) ssum[wid] = s;
  __syncthreads();
  if (threadIdx.x == 0) {
    float tot = 0.0f;
    for (int w = 0; w < nw; ++w) tot += ssum[w];
    sseP[blockIdx.x] = tot;
  }
}

// ---------------------------------------------------------------------------
// Reaction-time processing: bitonic sort 128 floats in LDS, jnp.quantile
// (linear) at q=0.25 / 0.75, clamp tails, renormalize mid-range.
// ---------------------------------------------------------------------------
__global__ void k_rt(const float* __restrict__ rt, float* __restrict__ rtF, int B) {
  __shared__ float s[128];
  __shared__ float smn[128], smx[128];
  int tid = threadIdx.x;
  float r = rt[tid];
  s[tid] = r;
  __syncthreads();
  for (int k = 2; k <= B; k <<= 1) {
    for (int j = k >> 1; j > 0; j >>= 1) {
      int ixj = tid ^ j;
      if (ixj > tid) {
        float a = s[tid], b = s[ixj];
        bool up = ((tid & k) == 0);
        if ((a > b) == up) { s[tid] = b; s[ixj] = a; }
      }
      __syncthreads();
    }
  }
  float pos25 = 0.25f * (float)(B - 1); int i25 = (int)pos25; float g25 = pos25 - (float)i25;
  float pos75 = 0.75f * (float)(B - 1); int i75 = (int)pos75; float g75 = pos75 - (float)i75;
  float lower = s[i25] * (1.0f - g25) + s[i25 + 1] * g25;
  float upper = s[i75] * (1.0f - g75) + s[i75 + 1] * g75;
  float rc = (r < lower) ? 0.0f : ((r > upper) ? 1.0f : r);
  bool mid = (rc != 0.0f) && (rc != 1.0f);
  smn[tid] = mid ? rc : 100.0f;
  smx[tid] = mid ? rc : -100.0f;
  __syncthreads();
  for (int st = B >> 1; st > 0; st >>= 1) {
    if (tid < st) {
      smn[tid] = fminf(smn[tid], smn[tid + st]);
      smx[tid] = fmaxf(smx[tid], smx[tid + st]);
    }
    __syncthreads();
  }
  float mn = smn[0], mx = smx[0];
  rtF[tid] = mid ? (rc - mn) / mx : rc;
}

// ---------------------------------------------------------------------------
// Per-row CE + first-index argmax; psych penalty on misclassification.
// Softmax-sum reduced with the WMMA wave sum.
// ---------------------------------------------------------------------------
__global__ void k_ce(const float* __restrict__ logits, const int* __restrict__ tgt,
                     const float* __restrict__ rtF, float* __restrict__ ceA, int C) {
  int row = blockIdx.x;
  const float* x = logits + (size_t)row * C;
  int tid = threadIdx.x;
  float bm = -INFINITY; int bi = 0x7fffffff;
  for (int c = tid; c < C; c += blockDim.x) {
    float v = x[c];
    if (v > bm || (v == bm && c < bi)) { bm = v; bi = c; }
  }
  for (int o = 16; o > 0; o >>= 1) {
    float ov = __shfl_xor(bm, o, 32);
    int   oi = __shfl_xor(bi, o, 32);
    if (ov > bm || (ov == bm && oi < bi)) { bm = ov; bi = oi; }
  }
  __shared__ float smv[8]; __shared__ int smi[8];
  __shared__ float srowmax; __shared__ int sargmax;
  int lane = tid & 31, wid = tid >> 5, nw = blockDim.x >> 5;
  if (lane == 0) { smv[wid] = bm; smi[wid] = bi; }
  __syncthreads();
  if (tid == 0) {
    float m = smv[0]; int mi = smi[0];
    for (int w = 1; w < nw; ++w)
      if (smv[w] > m || (smv[w] == m && smi[w] < mi)) { m = smv[w]; mi = smi[w]; }
    srowmax = m; sargmax = mi;
  }
  __syncthreads();
  float rowmax = srowmax;
  float ls = 0.0f;
  for (int c = tid; c < C; c += blockDim.x) ls += expf(x[c] - rowmax);
  float s = wave_sum(ls);   // v_wmma path
  __shared__ float ssum[8];
  if (lane == 0) ssum[wid] = s;
  __syncthreads();
  if (tid == 0) {
    float tot = 0.0f;
    for (int w = 0; w < nw; ++w) tot += ssum[w];
    int t = tgt[row];
    float ce = logf(tot) + rowmax - x[t];
    if (sargmax != t) ce += rtF[row];
    ceA[row] = ce;
  }
}

// Final combine: f64 accumulation of SSE partials and CE values.
__global__ void k_final(const float* __restrict__ sseP, int nSse,
                        const float* __restrict__ ceA, int B, int N1,
                        float* __restrict__ out) {
  __shared__ double sd[256];
  int tid = threadIdx.x;
  double a = 0.0;
  for (int i = tid; i < nSse; i += blockDim.x) a += (double)sseP[i];
  sd[tid] = a;
  __syncthreads();
  for (int st = blockDim.x >> 1; st > 0; st >>= 1) {
    if (tid < st) sd[tid] += sd[tid + st];
    __syncthreads();
  }
  double sse = sd[0];
  __syncthreads();
  double b = 0.0;
  for (int i = tid; i < B; i += blockDim.x) b += (double)ceA[i];
  sd[tid] = b;
  __syncthreads();
  for (int st = blockDim.x >> 1; st > 0; st >>= 1) {
    if (tid < st) sd[tid] += sd[tid + st];
    __syncthreads();
  }
  if (tid == 0) {
    double ceMean  = sd[0] / (double)B;
    double cyborg  = sse / (double)N1;
    out[0] = (float)(0.5 * ceMean + 0.5 * cyborg);
  }
}

// ---------------------------------------------------------------------------
extern "C" void kernel_launch(void* const* d_in, const int* in_sizes, int n_in,
                              void* d_out, int out_size, void* d_ws, size_t ws_size,
                              hipStream_t stream) {
  const float* logits = (const float*)d_in[0];   // [B,C] f32
  const int*   tgt    = (const int*)  d_in[1];   // [B] int32
  const float* rt     = (const float*)d_in[2];   // [B] f32
  const float* cams   = (const float*)d_in[3];   // [B,H,W] f32
  const float* ann    = (const float*)d_in[4];   // [B,3,H,W] f32

  const int B   = in_sizes[1];
  const int C   = in_sizes[0] / B;
  const int N1  = in_sizes[3];          // B*H*W
  const int HW  = N1 / B;
  const int hw4 = HW >> 2;
  const int n4  = N1 >> 2;

  const int NB = 1024, NT = 256;

  float* ws    = (float*)d_ws;
  float* cminP = ws;
  float* cmaxP = ws + NB;
  float* aminP = ws + 2 * NB;
  float* amaxP = ws + 3 * NB;
  float* scal  = ws + 4 * NB;   // 4 floats
  float* sseP  = scal + 4;      // NB floats
  float* rtF   = sseP + NB;     // B floats
  float* ceA   = rtF + B;       // B floats

  k_pass1 <<<NB, NT, 0, stream>>>((const float4*)cams, (const float4*)ann,
                                  cminP, cmaxP, aminP, amaxP, n4, hw4);
  k_reduce1<<<1,  NT, 0, stream>>>(cminP, cmaxP, aminP, amaxP, scal, NB);
  k_pass2 <<<NB, NT, 0, stream>>>((const float4*)cams, (const float4*)ann,
                                  scal, sseP, n4, hw4);
  k_rt    <<<1,  B,  0, stream>>>(rt, rtF, B);
  k_ce    <<<B,  NT, 0, stream>>>(logits, tgt, rtF, ceA, C);
  k_final <<<1,  NT, 0, stream>>>(sseP, NB, ceA, B, N1, (float*)d_out);
}